// StiefelDense_19911468384279
// MI455X (gfx1250) — compile-verified
//
#include <hip/hip_runtime.h>
#include <math.h>

// ---------------------------------------------------------------------------
// StiefelDense for MI455X (gfx1250, wave32).
//   1) power iteration (mat-vec, L2-resident)  -> sigma
//   2) theta0 = W / (sigma + 1e-5)
//   3) 100x Bjorck:  G = theta^T theta            (f32 WMMA GEMM)
//                    theta = 1.5*theta - 0.5*theta*G (f32 WMMA GEMM)
//   4) out = x @ theta + bias                     (f32 WMMA GEMM)
// GEMMs: V_WMMA_F32_16X16X4_F32 (fp32 matrix pipe, keeps reference fp32
// accuracy), double-buffered LDS tiles staged with GLOBAL_LOAD_ASYNC_TO_LDS
// (ASYNCcnt) -- builtin confirmed present on this toolchain.
// ---------------------------------------------------------------------------

typedef __attribute__((ext_vector_type(2))) float v2f;
typedef __attribute__((ext_vector_type(4))) float v4f;
typedef __attribute__((ext_vector_type(8))) float v8f;

#define DIN    2048
#define DOUT   2048
#define TOKENS 8192

// ---------------- async global->LDS staging helpers ----------------

#if __has_builtin(__builtin_amdgcn_global_load_async_to_lds_b128)
#define HAS_ASYNC_LDS 1
// Parameter types per compiler diagnostic:
//   param0: int __attribute__((vector_size(16))) AS1*   (global source)
//   param1: AS3 counterpart                              (LDS destination)
typedef int v4i_t __attribute__((vector_size(16)));
typedef __attribute__((address_space(1))) v4i_t* gv4i_p;
typedef __attribute__((address_space(3))) v4i_t* lv4i_p;
#else
#define HAS_ASYNC_LDS 0
#endif

static __device__ __forceinline__ void cp_async16(const float* g, float* l) {
#if HAS_ASYNC_LDS
  __builtin_amdgcn_global_load_async_to_lds_b128((gv4i_p)g, (lv4i_p)l, 0, 0);
#else
  *(v4f*)l = *(const v4f*)g;
#endif
}

template <int N>
static __device__ __forceinline__ void async_wait() {
#if HAS_ASYNC_LDS
  asm volatile("s_wait_asynccnt %0" ::"i"(N) : "memory");
#endif
}

// ---------------- vector / scalar helper kernels ----------------

__global__ void colmean_kernel(const float* __restrict__ W, float* __restrict__ u) {
  int j = blockIdx.x * blockDim.x + threadIdx.x;
  if (j >= DOUT) return;
  float s = 0.f;
  for (int i = 0; i < DIN; ++i) s += W[(size_t)i * DOUT + j];
  u[j] = s * (1.0f / (float)DIN);
}

// t = W @ u  (one block per row)
__global__ __launch_bounds__(256) void matvec_kernel(const float* __restrict__ W,
                                                     const float* __restrict__ u,
                                                     float* __restrict__ t) {
  __shared__ float red[256];
  const int row = blockIdx.x;
  float s = 0.f;
  for (int j = threadIdx.x; j < DOUT; j += 256)
    s += W[(size_t)row * DOUT + j] * u[j];
  red[threadIdx.x] = s;
  __syncthreads();
  for (int off = 128; off > 0; off >>= 1) {
    if (threadIdx.x < off) red[threadIdx.x] += red[threadIdx.x + off];
    __syncthreads();
  }
  if (threadIdx.x == 0) t[row] = red[0];
}

// out[j] = sum_i v[i] * W[i][j]   (thread per column, coalesced)
__global__ __launch_bounds__(256) void matvecT_kernel(const float* __restrict__ W,
                                                      const float* __restrict__ v,
                                                      float* __restrict__ out) {
  int j = blockIdx.x * 256 + threadIdx.x;
  if (j >= DOUT) return;
  float s = 0.f;
  for (int i = 0; i < DIN; ++i) s += v[i] * W[(size_t)i * DOUT + j];
  out[j] = s;
}

// v <- v / (||v|| + 1e-10), single block
__global__ __launch_bounds__(256) void normalize_kernel(float* __restrict__ v) {
  __shared__ float red[256];
  __shared__ float inv;
  float s = 0.f;
  for (int i = threadIdx.x; i < DOUT; i += 256) { float x = v[i]; s += x * x; }
  red[threadIdx.x] = s;
  __syncthreads();
  for (int off = 128; off > 0; off >>= 1) {
    if (threadIdx.x < off) red[threadIdx.x] += red[threadIdx.x + off];
    __syncthreads();
  }
  if (threadIdx.x == 0) inv = 1.0f / (sqrtf(red[0]) + 1e-10f);
  __syncthreads();
  for (int i = threadIdx.x; i < DOUT; i += 256) v[i] *= inv;
}

// sigma = v.(Wu) with v = t/(||t||+eps)  ==>  ||t||^2 / (||t|| + eps)
__global__ __launch_bounds__(256) void sigma_kernel(const float* __restrict__ t,
                                                    float* __restrict__ sig) {
  __shared__ float red[256];
  float s = 0.f;
  for (int i = threadIdx.x; i < DOUT; i += 256) { float x = t[i]; s += x * x; }
  red[threadIdx.x] = s;
  __syncthreads();
  for (int off = 128; off > 0; off >>= 1) {
    if (threadIdx.x < off) red[threadIdx.x] += red[threadIdx.x + off];
    __syncthreads();
  }
  if (threadIdx.x == 0) {
    float ss = red[0];
    sig[0] = ss / (sqrtf(ss) + 1e-10f);
  }
}

__global__ __launch_bounds__(256) void scale_matrix_kernel(const float* __restrict__ W,
                                                           const float* __restrict__ sig,
                                                           float* __restrict__ theta) {
  size_t i = (size_t)blockIdx.x * 256 + threadIdx.x;
  float inv = 1.0f / (sig[0] + 1e-5f);
  theta[i] = W[i] * inv;
}

// ---------------- f32 WMMA GEMM kernels ----------------
// Block tile 128x128, K-step 16, double-buffered LDS. 256 threads = 8 waves,
// wave tile 32x64 (2 M-subtiles x 4 N-subtiles of 16x16, v8f accum each).
// Fragment layouts per CDNA5 ISA (V_WMMA_F32_16X16X4_F32):
//   A 16x4 : lane L holds A[L&15][2*(L>>4) + {0,1}] in 2 VGPRs
//   B 4x16 : lane L holds B[2*(L>>4) + {0,1}][L&15] in 2 VGPRs

#define LDT 136  // stride for [16][128] K-major tiles (lane-half bank split)
#define LDA 20   // stride for [128][16] M-major tiles (16B-aligned rows)

static __device__ __forceinline__ v8f wmma_f32(v2f a, v2f b, v8f c) {
  return __builtin_amdgcn_wmma_f32_16x16x4_f32(false, a, false, b, (short)0, c,
                                               false, false);
}

// G = theta^T * theta  (both operand tiles are 16x128 row-chunks of theta)
__global__ __launch_bounds__(256) void gemm_ata_kernel(const float* __restrict__ T,
                                                       float* __restrict__ C) {
  __shared__ float Ts[2][16][LDT];  // theta[k][m_blk..+127]
  __shared__ float Us[2][16][LDT];  // theta[k][n_blk..+127]
  const int m_blk = blockIdx.x * 128;
  const int n_blk = blockIdx.y * 128;
  const int tid  = threadIdx.x;
  const int lane = tid & 31;
  const int wid  = tid >> 5;
  const int lh   = lane >> 4;  // lane half
  const int ll   = lane & 15;
  const int wm   = (wid & 3) * 32;
  const int wn   = (wid >> 2) * 64;

  v8f acc[2][4];
  const v8f z = {0.f, 0.f, 0.f, 0.f, 0.f, 0.f, 0.f, 0.f};
  for (int i = 0; i < 2; ++i)
    for (int j = 0; j < 4; ++j) acc[i][j] = z;

  // 16x128 tile: 512 float4s, 2 per thread per tile -> 4 async ops/thread
  auto load_tiles = [&](int b, int k0) {
    for (int p = 0; p < 2; ++p) {
      int q = tid + p * 256;
      int r = q >> 5, c = (q & 31) * 4;
      cp_async16(&T[(size_t)(k0 + r) * DOUT + m_blk + c], &Ts[b][r][c]);
      cp_async16(&T[(size_t)(k0 + r) * DOUT + n_blk + c], &Us[b][r][c]);
    }
  };

  const int S = DIN / 16;
  load_tiles(0, 0);
  for (int s = 0; s < S; ++s) {
    if (s + 1 < S) {
      load_tiles((s + 1) & 1, (s + 1) * 16);
      async_wait<4>();  // stage-s group complete (4 newest may be in flight)
    } else {
      async_wait<0>();
    }
    __syncthreads();  // all waves' stage-s tiles visible
    const int b = s & 1;
    for (int kk = 0; kk < 16; kk += 4) {
      const int r0 = kk + lh * 2;
      v2f af[2], bf[4];
      for (int i = 0; i < 2; ++i) {
        int m = wm + i * 16 + ll;
        af[i].x = Ts[b][r0][m];
        af[i].y = Ts[b][r0 + 1][m];
      }
      for (int j = 0; j < 4; ++j) {
        int n = wn + j * 16 + ll;
        bf[j].x = Us[b][r0][n];
        bf[j].y = Us[b][r0 + 1][n];
      }
      for (int i = 0; i < 2; ++i)
        for (int j = 0; j < 4; ++j) acc[i][j] = wmma_f32(af[i], bf[j], acc[i][j]);
    }
    __syncthreads();  // done reading buf b; next iter may overwrite it
  }
  for (int i = 0; i < 2; ++i)
    for (int j = 0; j < 4; ++j) {
      int mbase = m_blk + wm + i * 16 + lh * 8;
      int nbase = n_blk + wn + j * 16 + ll;
      for (int r = 0; r < 8; ++r)
        C[(size_t)(mbase + r) * DOUT + nbase] = acc[i][j][r];
    }
}

// MODE 0: D = 1.5*E - 0.5*(A*B)   (E = theta, same indices as D)  [Bjorck step]
// MODE 1: D = (A*B) + E[n]        (E = bias)                      [final x@theta]
template <int MODE>
__global__ __launch_bounds__(256) void gemm_nn_kernel(const float* __restrict__ A,
                                                      const float* __restrict__ B,
                                                      const float* __restrict__ E,
                                                      float* __restrict__ D) {
  __shared__ float As[2][128][LDA];  // A[m_blk..+127][k0..+15]
  __shared__ float Bs[2][16][LDT];   // B[k0..+15][n_blk..+127]
  const int m_blk = blockIdx.x * 128;
  const int n_blk = blockIdx.y * 128;
  const int tid  = threadIdx.x;
  const int lane = tid & 31;
  const int wid  = tid >> 5;
  const int lh   = lane >> 4;
  const int ll   = lane & 15;
  const int wm   = (wid & 3) * 32;
  const int wn   = (wid >> 2) * 64;

  v8f acc[2][4];
  const v8f z = {0.f, 0.f, 0.f, 0.f, 0.f, 0.f, 0.f, 0.f};
  for (int i = 0; i < 2; ++i)
    for (int j = 0; j < 4; ++j) acc[i][j] = z;

  // A tile 128x16 (512 float4) + B tile 16x128 (512 float4): 4 ops/thread
  auto load_tiles = [&](int b, int k0) {
    for (int p = 0; p < 2; ++p) {
      int q = tid + p * 256;
      int r = q >> 2, c = (q & 3) * 4;
      cp_async16(&A[(size_t)(m_blk + r) * DIN + k0 + c], &As[b][r][c]);
    }
    for (int p = 0; p < 2; ++p) {
      int q = tid + p * 256;
      int r = q >> 5, c = (q & 31) * 4;
      cp_async16(&B[(size_t)(k0 + r) * DOUT + n_blk + c], &Bs[b][r][c]);
    }
  };

  const int S = DIN / 16;
  load_tiles(0, 0);
  for (int s = 0; s < S; ++s) {
    if (s + 1 < S) {
      load_tiles((s + 1) & 1, (s + 1) * 16);
      async_wait<4>();
    } else {
      async_wait<0>();
    }
    __syncthreads();
    const int b = s & 1;
    for (int kk = 0; kk < 16; kk += 4) {
      const int r0 = kk + lh * 2;
      v2f af[2], bf[4];
      for (int i = 0; i < 2; ++i) {
        int m = wm + i * 16 + ll;
        af[i] = *(const v2f*)&As[b][m][r0];  // 8B aligned: r0 even, LDA even
      }
      for (int j = 0; j < 4; ++j) {
        int n = wn + j * 16 + ll;
        bf[j].x = Bs[b][r0][n];
        bf[j].y = Bs[b][r0 + 1][n];
      }
      for (int i = 0; i < 2; ++i)
        for (int j = 0; j < 4; ++j) acc[i][j] = wmma_f32(af[i], bf[j], acc[i][j]);
    }
    __syncthreads();
  }
  for (int i = 0; i < 2; ++i)
    for (int j = 0; j < 4; ++j) {
      int mbase = m_blk + wm + i * 16 + lh * 8;
      int nbase = n_blk + wn + j * 16 + ll;
      for (int r = 0; r < 8; ++r) {
        size_t idx = (size_t)(mbase + r) * DOUT + nbase;
        float v = acc[i][j][r];
        if (MODE == 0)
          D[idx] = 1.5f * E[idx] - 0.5f * v;
        else
          D[idx] = v + E[nbase];
      }
    }
}

// ---------------- host orchestration ----------------

extern "C" void kernel_launch(void* const* d_in, const int* in_sizes, int n_in,
                              void* d_out, int out_size, void* d_ws, size_t ws_size,
                              hipStream_t stream) {
  const float* x    = (const float*)d_in[0];  // 8192 x 2048
  const float* W    = (const float*)d_in[1];  // 2048 x 2048
  const float* bias = (const float*)d_in[2];  // 2048
  float* out = (float*)d_out;                 // 8192 x 2048

  // ws: theta_a (16MB) + u + t + sigma  (~16.02 MB needed)
  float* theta_a = (float*)d_ws;
  float* u   = theta_a + (size_t)DIN * DOUT;
  float* t   = u + DOUT;
  float* sig = t + DOUT;

  // d_out (64MB) doubles as Bjorck scratch; fully overwritten by final GEMM.
  float* Gbuf    = out;                           // Gram matrix, bytes [0,16MB)
  float* theta_b = out + (size_t)8 * 1024 * 1024; // bytes [32MB, 48MB)

  // ---- spectral normalization (power iteration, converged fixed point) ----
  colmean_kernel<<<DOUT / 256, 256, 0, stream>>>(W, u);
  normalize_kernel<<<1, 256, 0, stream>>>(u);
  for (int it = 0; it < 100; ++it) {
    matvec_kernel<<<DIN, 256, 0, stream>>>(W, u, t);          // t = W u
    normalize_kernel<<<1, 256, 0, stream>>>(t);               // v
    matvecT_kernel<<<DOUT / 256, 256, 0, stream>>>(W, t, u);  // u = v W
    normalize_kernel<<<1, 256, 0, stream>>>(u);
  }
  matvec_kernel<<<DIN, 256, 0, stream>>>(W, u, t);
  sigma_kernel<<<1, 256, 0, stream>>>(t, sig);
  scale_matrix_kernel<<<(DIN * DOUT) / 256, 256, 0, stream>>>(W, sig, theta_a);

  // ---- Bjorck orthogonalization: 100 iterations, 2 fp32 WMMA GEMMs each ----
  dim3 grid_sq(DIN / 128, DOUT / 128);
  float* cur = theta_a;
  float* nxt = theta_b;
  for (int it = 0; it < 100; ++it) {
    gemm_ata_kernel<<<grid_sq, 256, 0, stream>>>(cur, Gbuf);
    gemm_nn_kernel<0><<<grid_sq, 256, 0, stream>>>(cur, Gbuf, cur, nxt);
    float* tmp = cur; cur = nxt; nxt = tmp;
  }
  // even number of swaps: cur == theta_a (lives in d_ws, safe vs d_out write)

  // ---- out = x @ theta + bias ----
  dim3 grid_out(TOKENS / 128, DOUT / 128);
  gemm_nn_kernel<1><<<grid_out, 256, 0, stream>>>(x, cur, bias, out);
}